// FilterLayer_42314017800872
// MI455X (gfx1250) — compile-verified
//
#include <hip/hip_runtime.h>
#include <math.h>

#define B_TOTAL 1024
#define T_STEPS 128
#define DIM 24          // state dim
#define HID 32          // LSTM hidden
#define GATES 128       // 4*HID
#define DT_CONST 0.1f

typedef __attribute__((ext_vector_type(2))) float v2f;
typedef __attribute__((ext_vector_type(8))) float v8f;

__device__ __forceinline__ float sigmoidf_(float xv) {
    return 1.0f / (1.0f + __expf(-xv));
}

// ---------------------------------------------------------------------------
// Kernel 1: LSTM (PyTorch gate order i,f,g,o) over T steps + FC + abs.
// One workgroup = 16 batch rows, 128 threads (4 waves).
// Gate GEMM [16 x 56] @ [56 x 128] via v_wmma_f32_16x16x4_f32:
//   wave w computes gate-column tiles 2w and 2w+1.
// ---------------------------------------------------------------------------
__global__ __launch_bounds__(128) void lstm_fc_kernel(
    const float* __restrict__ x,      // [B, T, 24]
    const float* __restrict__ W_ih,   // [128, 24]
    const float* __restrict__ W_hh,   // [128, 32]
    const float* __restrict__ b_ih,   // [128]
    const float* __restrict__ b_hh,   // [128]
    const float* __restrict__ W_fc,   // [24, 32]
    const float* __restrict__ b_fc,   // [24]
    float* __restrict__ vec_out)      // [B, 24]
{
    __shared__ float sWih[GATES][DIM];
    __shared__ float sWhh[GATES][HID];
    __shared__ float sBias[GATES];
    __shared__ float sX[16][DIM];
    __shared__ float sH[16][HID];
    __shared__ float sC[16][HID];
    __shared__ float sG[16][GATES];

    const int tid  = threadIdx.x;
    const int lane = tid & 31;
    const int wave = tid >> 5;
    const int b0   = blockIdx.x * 16;

    for (int i = tid; i < GATES * DIM; i += 128) sWih[i / DIM][i % DIM] = W_ih[i];
    for (int i = tid; i < GATES * HID; i += 128) sWhh[i / HID][i % HID] = W_hh[i];
    for (int i = tid; i < GATES; i += 128)       sBias[i] = b_ih[i] + b_hh[i];
    for (int i = tid; i < 16 * HID; i += 128) {
        sH[i / HID][i % HID] = 0.0f;
        sC[i / HID][i % HID] = 0.0f;
    }
    __syncthreads();

    const int hi = lane >> 4;          // selects K-pair {2hi, 2hi+1} of each 4-slab
    const int nl = lane & 15;
    const int n0 = wave * 32 + nl;     // gate column, tile 2w
    const int n1 = n0 + 16;            // gate column, tile 2w+1

    for (int t = 0; t < T_STEPS; ++t) {
        // stage x[:, t, :] into LDS
        for (int i = tid; i < 16 * DIM; i += 128) {
            int m = i / DIM, d = i % DIM;
            sX[m][d] = x[(size_t)(b0 + m) * T_STEPS * DIM + (size_t)t * DIM + d];
        }
        __syncthreads();

        v8f acc0 = {0.f, 0.f, 0.f, 0.f, 0.f, 0.f, 0.f, 0.f};
        v8f acc1 = {0.f, 0.f, 0.f, 0.f, 0.f, 0.f, 0.f, 0.f};

        // input projection: K = 24 -> 6 slabs of 4
#pragma unroll
        for (int ks = 0; ks < 6; ++ks) {
            const int k0 = ks * 4 + hi * 2;
            v2f a;  a.x  = sX[nl][k0];    a.y  = sX[nl][k0 + 1];
            v2f bA; bA.x = sWih[n0][k0];  bA.y = sWih[n0][k0 + 1];
            v2f bB; bB.x = sWih[n1][k0];  bB.y = sWih[n1][k0 + 1];
            acc0 = __builtin_amdgcn_wmma_f32_16x16x4_f32(false, a, false, bA,
                                                         (short)0, acc0, false, false);
            acc1 = __builtin_amdgcn_wmma_f32_16x16x4_f32(false, a, false, bB,
                                                         (short)0, acc1, false, false);
        }
        // recurrent projection: K = 32 -> 8 slabs of 4
#pragma unroll
        for (int ks = 0; ks < 8; ++ks) {
            const int k0 = ks * 4 + hi * 2;
            v2f a;  a.x  = sH[nl][k0];    a.y  = sH[nl][k0 + 1];
            v2f bA; bA.x = sWhh[n0][k0];  bA.y = sWhh[n0][k0 + 1];
            v2f bB; bB.x = sWhh[n1][k0];  bB.y = sWhh[n1][k0 + 1];
            acc0 = __builtin_amdgcn_wmma_f32_16x16x4_f32(false, a, false, bA,
                                                         (short)0, acc0, false, false);
            acc1 = __builtin_amdgcn_wmma_f32_16x16x4_f32(false, a, false, bB,
                                                         (short)0, acc1, false, false);
        }
        // scatter D tiles (C/D layout: VGPR v -> row v + 8*hi, lane%16 -> col)
#pragma unroll
        for (int vv = 0; vv < 8; ++vv) {
            const int m = vv + 8 * hi;
            sG[m][n0] = acc0[vv];
            sG[m][n1] = acc1[vv];
        }
        __syncthreads();

        // LSTM cell activations
        for (int i = tid; i < 16 * HID; i += 128) {
            const int m = i >> 5, j = i & 31;
            const float iv = sigmoidf_(sG[m][j]        + sBias[j]);
            const float fv = sigmoidf_(sG[m][32 + j]   + sBias[32 + j]);
            const float gv = tanhf(    sG[m][64 + j]   + sBias[64 + j]);
            const float ov = sigmoidf_(sG[m][96 + j]   + sBias[96 + j]);
            const float cn = fv * sC[m][j] + iv * gv;
            sC[m][j] = cn;
            sH[m][j] = ov * tanhf(cn);
        }
        __syncthreads();
    }

    // FC + abs: vec[b, d] = |sum_j h[b,j] * W_fc[d,j] + b_fc[d]|
    for (int i = tid; i < 16 * DIM; i += 128) {
        const int m = i / DIM, d = i % DIM;
        float acc = b_fc[d];
#pragma unroll
        for (int j = 0; j < HID; ++j) acc += sH[m][j] * W_fc[d * HID + j];
        vec_out[(size_t)(b0 + m) * DIM + d] = fabsf(acc);
    }
}

// ---------------------------------------------------------------------------
// Kernel 2: Kalman filter scan, one workgroup per batch chain (128 threads).
// Per step: structured Pm = F P F^T + Q, Gauss-Jordan solve of [S | Pm],
// state update, and P = Pm - Kt^T * Pm via padded 32x32 WMMA (one 16x16
// output tile per wave, K=24 -> 6 slabs).
// ---------------------------------------------------------------------------
__global__ __launch_bounds__(128) void kalman_kernel(
    const float* __restrict__ x,     // [B, T, 24]
    const float* __restrict__ qvec,  // [B, 24]
    const float* __restrict__ rvec,  // [B, 24]
    float* __restrict__ out)         // [B, T, 24]
{
    __shared__ float sP [DIM][DIM];
    __shared__ float sPm[DIM][DIM];
    __shared__ float sM [DIM][2 * DIM];   // augmented [S | Pm] -> [S^-1 S | Kt]
    __shared__ float spiv[2 * DIM];
    __shared__ float sfac[DIM];
    __shared__ float ss[DIM], ssm[DIM], sy[DIM], sq[DIM], sr[DIM], sz[DIM];

    const int tid  = threadIdx.x;
    const int lane = tid & 31;
    const int wave = tid >> 5;
    const int b    = blockIdx.x;
    const size_t base = (size_t)b * T_STEPS * DIM;

    if (tid < DIM) {
        ss[tid] = x[base + tid];            // s0 = x[:, 0]
        sq[tid] = qvec[b * DIM + tid];
        sr[tid] = rvec[b * DIM + tid];
    }
    for (int i = tid; i < DIM * DIM; i += 128)
        sP[i / DIM][i % DIM] = (i / DIM == i % DIM) ? 1.0f : 0.0f;   // P0 = I
    __syncthreads();

    const int hi = lane >> 4;
    const int nl = lane & 15;
    const int tm = wave >> 1, tn = wave & 1;  // 2x2 tile grid over padded 32x32
    const int n  = tn * 16 + nl;              // output column (B/C/D)
    const int ma = tm * 16 + nl;              // A-matrix row

    for (int t = 0; t < T_STEPS; ++t) {
        if (tid < DIM) sz[tid] = x[base + (size_t)t * DIM + tid];
        __syncthreads();

        // predict state: s_m = F s   (F = kron(I6, [[1,0,dt,0],[0,1,0,dt],[0,0,1,0],[0,0,0,1]]))
        if (tid < DIM) {
            const int i = tid;
            const float sm = ss[i] + (((i & 3) < 2) ? DT_CONST * ss[i + 2] : 0.0f);
            ssm[i] = sm;
            sy[i]  = sz[i] - sm;
        }
        // predict covariance: Pm = F P F^T + diag(q), exploiting F structure
        for (int idx = tid; idx < DIM * DIM; idx += 128) {
            const int i = idx / DIM, j = idx % DIM;
            const bool ai = (i & 3) < 2, aj = (j & 3) < 2;
            float vv = sP[i][j];
            if (ai)       vv += DT_CONST * sP[i + 2][j];
            if (aj)       vv += DT_CONST * sP[i][j + 2];
            if (ai && aj) vv += DT_CONST * DT_CONST * sP[i + 2][j + 2];
            if (i == j)   vv += sq[i];
            sPm[i][j] = vv;
        }
        __syncthreads();

        // augmented system [S | Pm], S = Pm + diag(r)  (Pm symmetric => Pm^T = Pm)
        for (int idx = tid; idx < DIM * 2 * DIM; idx += 128) {
            const int i = idx / (2 * DIM), j = idx % (2 * DIM);
            sM[i][j] = (j < DIM) ? (sPm[i][j] + ((i == j) ? sr[i] : 0.0f))
                                 : sPm[i][j - DIM];
        }
        __syncthreads();

        // Gauss-Jordan: right half becomes Kt = S^-1 Pm
        for (int p = 0; p < DIM; ++p) {
            const float pinv = 1.0f / sM[p][p];
            for (int j = tid; j < 2 * DIM; j += 128) spiv[j] = sM[p][j] * pinv;
            if (tid < DIM) sfac[tid] = sM[tid][p];
            __syncthreads();
            for (int idx = tid; idx < DIM * 2 * DIM; idx += 128) {
                const int i = idx / (2 * DIM), j = idx % (2 * DIM);
                sM[i][j] = (i == p) ? spiv[j] : sM[i][j] - sfac[i] * spiv[j];
            }
            __syncthreads();
        }

        // state update: s = s_m + K y,  K[i][j] = Kt[j][i] = sM[j][24+i]
        if (tid < DIM) {
            const int i = tid;
            float acc = ssm[i];
#pragma unroll
            for (int j = 0; j < DIM; ++j) acc += sM[j][DIM + i] * sy[j];
            ss[i] = acc;
            out[base + (size_t)t * DIM + i] = acc;
        }

        // covariance update: P = Pm - Kt^T Pm  (WMMA: D = (-Kt^T) * Pm + Pm)
        {
            v8f c;
#pragma unroll
            for (int vv = 0; vv < 8; ++vv) {
                const int m = tm * 16 + vv + 8 * hi;
                c[vv] = (m < DIM && n < DIM) ? sPm[m][n] : 0.0f;
            }
#pragma unroll
            for (int ks = 0; ks < 6; ++ks) {
                const int k0 = ks * 4 + hi * 2;
                v2f a;
                a.x = (ma < DIM) ? -sM[k0][DIM + ma]     : 0.0f;
                a.y = (ma < DIM) ? -sM[k0 + 1][DIM + ma] : 0.0f;
                v2f bb;
                bb.x = (n < DIM) ? sPm[k0][n]     : 0.0f;
                bb.y = (n < DIM) ? sPm[k0 + 1][n] : 0.0f;
                c = __builtin_amdgcn_wmma_f32_16x16x4_f32(false, a, false, bb,
                                                          (short)0, c, false, false);
            }
#pragma unroll
            for (int vv = 0; vv < 8; ++vv) {
                const int m = tm * 16 + vv + 8 * hi;
                if (m < DIM && n < DIM) sP[m][n] = c[vv];
            }
        }
        __syncthreads();
    }
}

// ---------------------------------------------------------------------------
extern "C" void kernel_launch(void* const* d_in, const int* in_sizes, int n_in,
                              void* d_out, int out_size, void* d_ws, size_t ws_size,
                              hipStream_t stream) {
    const float* x      = (const float*)d_in[0];
    const float* q_W_ih = (const float*)d_in[1];
    const float* q_W_hh = (const float*)d_in[2];
    const float* q_b_ih = (const float*)d_in[3];
    const float* q_b_hh = (const float*)d_in[4];
    const float* q_W_fc = (const float*)d_in[5];
    const float* q_b_fc = (const float*)d_in[6];
    const float* r_W_ih = (const float*)d_in[7];
    const float* r_W_hh = (const float*)d_in[8];
    const float* r_b_ih = (const float*)d_in[9];
    const float* r_b_hh = (const float*)d_in[10];
    const float* r_W_fc = (const float*)d_in[11];
    const float* r_b_fc = (const float*)d_in[12];

    float* qvec = (float*)d_ws;                    // [B, 24]
    float* rvec = qvec + (size_t)B_TOTAL * DIM;    // [B, 24]
    float* out  = (float*)d_out;                   // [B, T, 24]

    lstm_fc_kernel<<<B_TOTAL / 16, 128, 0, stream>>>(
        x, q_W_ih, q_W_hh, q_b_ih, q_b_hh, q_W_fc, q_b_fc, qvec);
    lstm_fc_kernel<<<B_TOTAL / 16, 128, 0, stream>>>(
        x, r_W_ih, r_W_hh, r_b_ih, r_b_hh, r_W_fc, r_b_fc, rvec);
    kalman_kernel<<<B_TOTAL, 128, 0, stream>>>(x, qvec, rvec, out);
}